// cppSlideLayer_352187319095
// MI455X (gfx1250) — compile-verified
//
#include <hip/hip_runtime.h>

// CDNA5 wave32 WMMA types
typedef __attribute__((ext_vector_type(2))) float v2f;
typedef __attribute__((ext_vector_type(8))) float v8f;

#define SLIDE_IN_DIM   16384
#define SLIDE_K_IN     1024
#define SLIDE_K_OUT    1024
#define WG_THREADS     256     // 8 wave32
#define J_PER_WG       128     // 8 waves x 16 outputs

// out[b, j] = sum_i W[out_idx[b,j], in_idx[b,i]] * v[b,i] + bias[out_idx[b,j]]
//
// One workgroup = (batch b, chunk of 128 j's). Each wave32 owns a 16-j tile and
// accumulates the K=1024 reduction with V_WMMA_F32_16X16X4_F32:
//   A (16x4 f32): lane m (0-15 / 16-31 halves) holds gathered W[row_m][col_{k}]
//                 for k = 2*half + {0,1} in VGPR {0,1}          (ISA 7.12.2)
//   B (4x16 f32): v[k] broadcast across all N columns (same (vgpr,half)->k map)
//   D (16x16 f32): every N column identical; lanes 0/16 hold D[0..7][0]/D[8..15][0]
__global__ __launch_bounds__(WG_THREADS)
void slide_wmma_gather_kernel(const float* __restrict__ in_values,
                              const float* __restrict__ W,
                              const float* __restrict__ bias,
                              const int*   __restrict__ in_idx,
                              const int*   __restrict__ out_idx,
                              float*       __restrict__ out)
{
    __shared__ int2  s_pair[SLIDE_K_IN];   // (col, val bits) interleaved -> 1x ds_load_b128 / wmma
    __shared__ int   s_row[J_PER_WG];      // gathered output rows
    __shared__ float s_bias[J_PER_WG];     // pre-gathered bias

    const int b     = blockIdx.y;
    const int jbase = blockIdx.x * J_PER_WG;
    const int tid   = threadIdx.x;

    // ---- cooperative staging of the per-batch reused data ----
    const int*   __restrict__ bcol = in_idx    + b * SLIDE_K_IN;
    const float* __restrict__ bval = in_values + b * SLIDE_K_IN;
#pragma unroll
    for (int k = tid; k < SLIDE_K_IN; k += WG_THREADS) {
        s_pair[k] = make_int2(bcol[k], __float_as_int(bval[k]));
    }
    if (tid < J_PER_WG) {
        const int r = out_idx[b * SLIDE_K_OUT + jbase + tid];
        s_row[tid]  = r;
        s_bias[tid] = bias[r];
    }
    __syncthreads();

    // ---- per-wave 16-output tile ----
    const int wave = tid >> 5;
    const int lane = tid & 31;
    const int m    = lane & 15;       // M index within tile
    const int half = lane >> 4;       // 0: K={0,1}, 1: K={2,3} per ISA layout

    const int jt      = wave * 16;                 // tile offset within chunk
    const int row     = s_row[jt + m];
    const int rowbase = row << 14;                 // * IN_DIM (elements; <2^28)

    v8f acc0 = {};
    v8f acc1 = {};

    // K loop, 8 per iteration (two independent WMMA accumulator chains)
    for (int i = 0; i < SLIDE_K_IN; i += 8) {
        const int k0 = i + 2 * half;               // this (vgpr,half) K slot
        const int k1 = k0 + 4;

        // broadcast ds_load_b128: {col[k], val[k], col[k+1], val[k+1]}
        const int4 p0 = *reinterpret_cast<const int4*>(&s_pair[k0]);
        const int4 p1 = *reinterpret_cast<const int4*>(&s_pair[k1]);

        // the irreducible random gather from W (global_load_b32 x4 per lane)
        const float a00 = W[rowbase + p0.x];
        const float a01 = W[rowbase + p0.z];
        const float a10 = W[rowbase + p1.x];
        const float a11 = W[rowbase + p1.z];

        const v2f A0 = {a00, a01};
        const v2f B0 = {__int_as_float(p0.y), __int_as_float(p0.w)};
        const v2f A1 = {a10, a11};
        const v2f B1 = {__int_as_float(p1.y), __int_as_float(p1.w)};

        acc0 = __builtin_amdgcn_wmma_f32_16x16x4_f32(
                   false, A0, false, B0, (short)0, acc0, false, false);
        acc1 = __builtin_amdgcn_wmma_f32_16x16x4_f32(
                   false, A1, false, B1, (short)0, acc1, false, false);
    }

    const v8f acc = acc0 + acc1;

    // D extraction: every N column identical; lane 0 holds M=0..7 in its 8
    // accumulator VGPRs, lane 16 holds M=8..15.
    if (m == 0) {
        const int   mb = half * 8;
        float*       __restrict__ op = out + b * SLIDE_K_OUT + jbase + jt + mb;
        const float* __restrict__ bp = s_bias + jt + mb;
#pragma unroll
        for (int r = 0; r < 8; ++r) {
            op[r] = acc[r] + bp[r];
        }
    }
}

extern "C" void kernel_launch(void* const* d_in, const int* in_sizes, int n_in,
                              void* d_out, int out_size, void* d_ws, size_t ws_size,
                              hipStream_t stream)
{
    (void)n_in; (void)out_size; (void)d_ws; (void)ws_size;

    const float* in_values = (const float*)d_in[0];   // [B, K_IN] f32
    const float* weights   = (const float*)d_in[1];   // [OUT_DIM, IN_DIM] f32
    const float* bias      = (const float*)d_in[2];   // [OUT_DIM] f32
    const int*   ain       = (const int*)  d_in[3];   // [B, K_IN] i32
    const int*   aout      = (const int*)  d_in[4];   // [B, K_OUT] i32
    float*       out       = (float*)d_out;           // [B, K_OUT] f32

    const int batch = in_sizes[0] / SLIDE_K_IN;       // 64

    dim3 grid(SLIDE_K_OUT / J_PER_WG, batch);         // (8, 64)
    slide_wmma_gather_kernel<<<grid, WG_THREADS, 0, stream>>>(
        in_values, weights, bias, ain, aout, out);
}